// Attn_3384434229614
// MI455X (gfx1250) — compile-verified
//
#include <hip/hip_runtime.h>
#include <math.h>

// Problem sizes (fixed by the reference)
#define SEQ  2048
#define BAT  32
#define HID  1024

typedef __attribute__((ext_vector_type(2))) float v2f;
typedef __attribute__((ext_vector_type(8))) float v8f;

// ---------------------------------------------------------------------------
// Kernel 1: v[b,h] = sum_o hidden[b,o] * W[o,h]   via V_WMMA_F32_16X16X4_F32
//   GEMM: M=32 (b), N=1024 (h), K=1024 (o). fp32 WMMA is numerically exact
//   fp32 FMA, which the sharp softmax (sigma(E)~32) requires.
//   One wave (32 threads) per 16x16 C tile; 2 x 64 = 128 tiles.
// ---------------------------------------------------------------------------
__global__ __launch_bounds__(32) void proj_v_wmma(const float* __restrict__ hidden,
                                                  const float* __restrict__ W,
                                                  float* __restrict__ vout) {
    const int lane  = threadIdx.x;        // 0..31, wave32
    const int lo    = lane & 15;
    const int hi    = lane >> 4;          // 0 or 1
    const int tileM = blockIdx.x & 1;     // 32 rows  -> 2 tiles
    const int tileN = blockIdx.x >> 1;    // 1024 cols -> 64 tiles

    const int m = tileM * 16 + lo;        // A row this lane supplies
    const int n = tileN * 16 + lo;        // B col this lane supplies

    const float* __restrict__ Arow = hidden + (size_t)m * HID;

    v8f acc = {};
    #pragma unroll 4
    for (int k = 0; k < HID; k += 4) {
        const int kk = k + 2 * hi;                      // even -> 8B aligned
        // A 16x4 layout (ISA 7.12.2): vgpr0=K(2*hi), vgpr1=K(2*hi+1)
        v2f a = *(const v2f*)(Arow + kk);
        // B 4x16 layout: vgpr v holds row (v + 2*hi), col = lane&15
        v2f b;
        b.x = W[(size_t)kk * HID + n];
        b.y = W[(size_t)(kk + 1) * HID + n];
        acc = __builtin_amdgcn_wmma_f32_16x16x4_f32(
            /*neg_a=*/false, a, /*neg_b=*/false, b,
            /*c_mod=*/(short)0, acc, /*reuse_a=*/false, /*reuse_b=*/false);
    }

    // C 16x16 layout: vgpr r -> row r (lanes 0-15) / row r+8 (lanes 16-31)
    #pragma unroll
    for (int r = 0; r < 8; ++r) {
        const int row = tileM * 16 + r + 8 * hi;
        vout[(size_t)row * HID + tileN * 16 + lo] = acc[r];
    }
}

// ---------------------------------------------------------------------------
// Kernel 2: c[b] = hidden[b,:] . bias[:]   (tiny; bias is zeros in setup but
// must be honored). One 256-thread block per b, LDS tree reduction.
// ---------------------------------------------------------------------------
__global__ __launch_bounds__(256) void bias_dot(const float* __restrict__ hidden,
                                                const float* __restrict__ bias,
                                                float* __restrict__ c) {
    __shared__ float red[256];
    const int b = blockIdx.x;
    const int t = threadIdx.x;
    float acc = 0.f;
    #pragma unroll
    for (int k = 0; k < HID / 256; ++k) {
        const int idx = t + k * 256;
        acc += hidden[(size_t)b * HID + idx] * bias[idx];
    }
    red[t] = acc;
    __syncthreads();
    for (int s = 128; s > 0; s >>= 1) {
        if (t < s) red[t] += red[t + s];
        __syncthreads();
    }
    if (t == 0) c[b] = red[0];
}

// ---------------------------------------------------------------------------
// Kernel 3: E[b,s] = enc[s,b,:] . v[b,:] + c[b]   -- the 256 MB stream.
// grid = (B, 64 s-chunks), block = 256 (8 waves). Each wave: fixed b, 4 s.
// v[b] lives in registers (8 x float4 per lane), enc read as b128 loads.
// ---------------------------------------------------------------------------
__global__ __launch_bounds__(256) void energies_kernel(const float* __restrict__ enc,
                                                       const float* __restrict__ v,
                                                       const float* __restrict__ c,
                                                       float* __restrict__ out) {
    const int b    = blockIdx.x;
    const int wave = threadIdx.x >> 5;
    const int lane = threadIdx.x & 31;
    const int s0   = blockIdx.y * 32 + wave * 4;

    // Load this lane's 32 floats of v[b] (coalesced float4s, stays in VGPRs)
    const float4* __restrict__ v4 = (const float4*)(v + (size_t)b * HID);
    float4 vv[8];
    #pragma unroll
    for (int j = 0; j < 8; ++j) vv[j] = v4[j * 32 + lane];

    const float cb = c[b];

    #pragma unroll
    for (int i = 0; i < 4; ++i) {
        const int s = s0 + i;
        const float4* __restrict__ e4 =
            (const float4*)(enc + ((size_t)s * BAT + b) * HID);
        float acc = 0.f;
        #pragma unroll
        for (int j = 0; j < 8; ++j) {
            const float4 e = e4[j * 32 + lane];
            acc += e.x * vv[j].x + e.y * vv[j].y + e.z * vv[j].z + e.w * vv[j].w;
        }
        // wave32 xor-shuffle reduction
        #pragma unroll
        for (int off = 16; off >= 1; off >>= 1)
            acc += __shfl_xor(acc, off, 32);
        if (lane == 0) out[(size_t)b * SEQ + s] = acc + cb;
    }
}

// ---------------------------------------------------------------------------
// Kernel 4: in-place softmax over each row of out[B, S].
// ---------------------------------------------------------------------------
__global__ __launch_bounds__(256) void softmax_rows(float* __restrict__ out) {
    __shared__ float red[256];
    const int b = blockIdx.x;
    const int t = threadIdx.x;
    float* __restrict__ row = out + (size_t)b * SEQ;

    float x[SEQ / 256];
    float m = -INFINITY;
    #pragma unroll
    for (int k = 0; k < SEQ / 256; ++k) {
        x[k] = row[t + k * 256];
        m = fmaxf(m, x[k]);
    }
    red[t] = m;
    __syncthreads();
    for (int s = 128; s > 0; s >>= 1) {
        if (t < s) red[t] = fmaxf(red[t], red[t + s]);
        __syncthreads();
    }
    m = red[0];
    __syncthreads();

    float sum = 0.f;
    #pragma unroll
    for (int k = 0; k < SEQ / 256; ++k) {
        x[k] = expf(x[k] - m);
        sum += x[k];
    }
    red[t] = sum;
    __syncthreads();
    for (int s = 128; s > 0; s >>= 1) {
        if (t < s) red[t] += red[t + s];
        __syncthreads();
    }
    const float inv = 1.f / red[0];
    #pragma unroll
    for (int k = 0; k < SEQ / 256; ++k) row[t + k * 256] = x[k] * inv;
}

// ---------------------------------------------------------------------------
extern "C" void kernel_launch(void* const* d_in, const int* in_sizes, int n_in,
                              void* d_out, int out_size, void* d_ws, size_t ws_size,
                              hipStream_t stream) {
    (void)in_sizes; (void)n_in; (void)out_size; (void)ws_size;

    const float* hidden = (const float*)d_in[0];  // [1,B,H]
    const float* enc    = (const float*)d_in[1];  // [S,B,H]
    const float* W      = (const float*)d_in[2];  // [H,H]
    const float* bias   = (const float*)d_in[3];  // [H]
    float*       out    = (float*)d_out;          // [B,1,S] == [B,S]

    float* v_ws = (float*)d_ws;                   // [B,H] = 32768 floats
    float* c_ws = v_ws + (size_t)BAT * HID;       // [B]   = 32 floats

    // v = hidden @ W  (WMMA fp32, 128 one-wave tiles)
    proj_v_wmma<<<128, 32, 0, stream>>>(hidden, W, v_ws);
    // c[b] = hidden[b] . bias
    bias_dot<<<BAT, 256, 0, stream>>>(hidden, bias, c_ws);
    // E[b,s] = enc[s,b] . v[b] + c[b]   (streams 256 MB, HBM-bound)
    energies_kernel<<<dim3(BAT, SEQ / 32), 256, 0, stream>>>(enc, v_ws, c_ws, out);
    // softmax over s, in place
    softmax_rows<<<BAT, 256, 0, stream>>>(out);
}